// PixelLevelMatching_12678743458039
// MI455X (gfx1250) — compile-verified
//
#include <hip/hip_runtime.h>
#include <hip/hip_bf16.h>

#define BB     4
#define CIN    512
#define COUT   128
#define HH     64
#define WW     64
#define HW     (HH * WW)      // 4096 pixels per sample
#define NTPL   4096           // templates per sample

typedef __attribute__((ext_vector_type(2))) float v2f;
typedef __attribute__((ext_vector_type(8))) float v8f;
typedef __attribute__((ext_vector_type(4))) int   v4i;

// ---------------------------------------------------------------------------
// CDNA5 async global->LDS staging (ASYNCcnt path), with sync fallback.
// ---------------------------------------------------------------------------
#if defined(__gfx1250__) && \
    __has_builtin(__builtin_amdgcn_global_load_async_to_lds_b128) && \
    __has_builtin(__builtin_amdgcn_s_wait_asynccnt)
#define PLM_ASYNC 1
#else
#define PLM_ASYNC 0
#endif

// Copy one 16x128-f32 template tile (8 KB) into LDS; 256 threads, 32 B each.
__device__ __forceinline__ void tile_copy(float* dst_lds, const float* src, int tid) {
#if PLM_ASYNC
    __attribute__((address_space(1))) v4i* g =
        (__attribute__((address_space(1))) v4i*)(src) + tid;      // 16 B / thread / op
    __attribute__((address_space(3))) v4i* l =
        (__attribute__((address_space(3))) v4i*)(dst_lds) + tid;
    __builtin_amdgcn_global_load_async_to_lds_b128(g,       l,       0, 0);
    __builtin_amdgcn_global_load_async_to_lds_b128(g + 256, l + 256, 0, 0);  // +4 KB
#else
    const float4* s4 = (const float4*)src;
    float4*       d4 = (float4*)dst_lds;
    d4[tid]       = s4[tid];
    d4[tid + 256] = s4[tid + 256];
#endif
}

__device__ __forceinline__ void tile_wait() {
#if PLM_ASYNC
    __builtin_amdgcn_s_wait_asynccnt(0);
#endif
}

// ---------------------------------------------------------------------------
// Kernel 1: 1x1 conv as fp32 WMMA GEMM, feats read exactly once.
//   One wave per (b, 16-pixel tile); loops over all 8 o-tiles internally
//   (64 accumulator VGPRs), so the 33.5 MB feats tensor is single-pass and
//   the 256 KB weight matrix is the cache-resident operand.
// Output TRANSPOSED to convT[b][p][o] for the correlation stage.
// ---------------------------------------------------------------------------
__global__ __launch_bounds__(32)
void plm_conv_gemm(const float* __restrict__ feats,   // [B, CIN, HW]
                   const float* __restrict__ W,       // [COUT, CIN]
                   const float* __restrict__ bias,    // [COUT]
                   float* __restrict__ convT)         // [B, HW, COUT]
{
    const int pt   = blockIdx.x;          // pixel tile (0..255)
    const int b    = blockIdx.y;
    const int lane = threadIdx.x;
    const int half = lane >> 4;           // selects K pair {0,1} vs {2,3}
    const int l15  = lane & 15;
    const int p0   = pt * 16;

    // B column for this lane: feats[b][*][p0 + l15]  (K strided by HW)
    const float* fB = feats + (size_t)b * CIN * HW + p0 + l15;
    // A row base for this lane (o-tile 0): W[l15][*]
    const float* wA = W + (size_t)l15 * CIN + 2 * half;

    v8f acc[8];
#pragma unroll
    for (int ot = 0; ot < 8; ++ot) acc[ot] = (v8f){};

    for (int k = 0; k < CIN; k += 4) {
        const float* fp = fB + (size_t)(k + 2 * half) * HW;
        v2f bf;
        bf.x = fp[0];                                   // B[k+2h  ][n]
        bf.y = fp[HW];                                  // B[k+2h+1][n]
        const float* wrow = wA + k;
#pragma unroll
        for (int ot = 0; ot < 8; ++ot) {
            v2f a = *(const v2f*)(wrow + (size_t)ot * 16 * CIN);
            acc[ot] = __builtin_amdgcn_wmma_f32_16x16x4_f32(
                          false, a, false, bf, (short)0, acc[ot], false, false);
        }
    }

    // D layout: VGPR r, lanes 0-15 -> M=r, lanes 16-31 -> M=r+8; N = l15.
    float* orow = convT + ((size_t)b * HW + p0 + l15) * COUT + 8 * half;
#pragma unroll
    for (int ot = 0; ot < 8; ++ot) {
#pragma unroll
        for (int r = 0; r < 8; ++r)
            orow[ot * 16 + r] = acc[ot][r] + bias[ot * 16 + 8 * half + r];
    }
}

// ---------------------------------------------------------------------------
// Kernel 2: per-pixel channel L2 normalization (one wave per pixel, each lane
// owns 4 channels), in-place on convT, scatter to out_norm [B, COUT, HW],
// and fused pm_int dot products -> int_scores [B, 2, HW].
// ---------------------------------------------------------------------------
__global__ __launch_bounds__(256)
void plm_normalize(float* __restrict__ convT,            // [B, HW, COUT] in/out
                   const float* __restrict__ pm_int_bg,  // [B, COUT]
                   const float* __restrict__ pm_int_fg,  // [B, COUT]
                   float* __restrict__ out_norm,         // [B, COUT, HW]
                   float* __restrict__ out_int)          // [B, 2, HW]
{
    const int wid  = (int)((blockIdx.x * blockDim.x + threadIdx.x) >> 5); // pixel id
    const int lane = threadIdx.x & 31;
    const int b    = wid >> 12;          // / HW
    const int p    = wid & (HW - 1);

    float* row = convT + (size_t)wid * COUT;
    float4 v = *(const float4*)(row + lane * 4);

    float ss = v.x * v.x + v.y * v.y + v.z * v.z + v.w * v.w;
#pragma unroll
    for (int off = 16; off > 0; off >>= 1)
        ss += __shfl_xor(ss, off, 32);

    const float inv = 1.0f / sqrtf(ss);
    v.x *= inv; v.y *= inv; v.z *= inv; v.w *= inv;
    *(float4*)(row + lane * 4) = v;

    float* on = out_norm + ((size_t)b * COUT + lane * 4) * HW + p;
    on[0 * HW] = v.x; on[1 * HW] = v.y; on[2 * HW] = v.z; on[3 * HW] = v.w;

    const float4 tb = *(const float4*)(pm_int_bg + (size_t)b * COUT + lane * 4);
    const float4 tf = *(const float4*)(pm_int_fg + (size_t)b * COUT + lane * 4);
    float db = v.x * tb.x + v.y * tb.y + v.z * tb.z + v.w * tb.w;
    float df = v.x * tf.x + v.y * tf.y + v.z * tf.z + v.w * tf.w;
#pragma unroll
    for (int off = 16; off > 0; off >>= 1) {
        db += __shfl_xor(db, off, 32);
        df += __shfl_xor(df, off, 32);
    }
    if (lane == 0) {
        out_int[((size_t)b * 2 + 0) * HW + p] = db;
        out_int[((size_t)b * 2 + 1) * HW + p] = df;
    }
}

// ---------------------------------------------------------------------------
// Kernel 3: correlation GEMM + fused row-max via fp32 WMMA.
//   scores[p, n] = sum_c normP[p, c] * T[n, c];  out[p] = max_n scores[p, n]
// Workgroup = 8 waves x 32 pixels = 256 pixels. Each 16x128 template tile
// (8 KB) is staged into LDS ONCE (double-buffered, async-to-LDS path) and
// consumed by all 8 waves; each wave holds two 16x128 A tiles in registers so
// every B fragment feeds two WMMAs. Template L2 traffic drops 256x -> ~0.5 GB.
// ---------------------------------------------------------------------------
__global__ __launch_bounds__(256)
void plm_corr_max(const float* __restrict__ convT,  // [B, HW, COUT] normalized
                  const float* __restrict__ t0,     // pm_init_bg [B, N, COUT]
                  const float* __restrict__ t1,     // pm_init_fg
                  const float* __restrict__ t2,     // pm_prev_bg
                  const float* __restrict__ t3,     // pm_prev_fg
                  float* __restrict__ out)          // init @0, prev @ B*2*HW
{
    __shared__ float ldsT[2][16 * COUT];            // 2 x 8 KB double buffer

    const int wg   = blockIdx.x;                    // 0..15 (256 pixels each)
    const int set  = blockIdx.y;                    // 0..3
    const int b    = blockIdx.z;
    const int tid  = threadIdx.x;
    const int wave = tid >> 5;
    const int lane = tid & 31;
    const int half = lane >> 4;
    const int l15  = lane & 15;

    const float* Tset = (set == 0) ? t0 : (set == 1) ? t1 : (set == 2) ? t2 : t3;
    const float* T = Tset + (size_t)b * NTPL * COUT;
    const int p0 = wg * 256 + wave * 32;            // this wave's 32 pixels

    // Two 16x128 A tiles in registers (lane row = p0 [+16] + l15).
    v2f a0[32], a1[32];
    const float* pr = convT + ((size_t)b * HW + p0 + l15) * COUT + 2 * half;
#pragma unroll
    for (int kk = 0; kk < 32; ++kk) {
        a0[kk] = *(const v2f*)(pr + 4 * kk);
        a1[kk] = *(const v2f*)(pr + 16 * COUT + 4 * kk);
    }

    v8f m0, m1;
#pragma unroll
    for (int r = 0; r < 8; ++r) { m0[r] = -3.402823466e+38f; m1[r] = m0[r]; }

    tile_copy(ldsT[0], T, tid);                     // prologue: tile 0

    for (int nt = 0; nt < NTPL / 16; ++nt) {
        const int cur = nt & 1;
        tile_wait();                                 // my async stores landed
        __syncthreads();                             // tile[cur] visible; tile[cur^1] free
        if (nt + 1 < NTPL / 16)
            tile_copy(ldsT[cur ^ 1], T + (size_t)(nt + 1) * 16 * COUT, tid);

        const float* tb = &ldsT[cur][l15 * COUT + 2 * half];
        v8f c0 = {}, c1 = {};
#pragma unroll
        for (int kk = 0; kk < 32; ++kk) {
            v2f bf = *(const v2f*)(tb + 4 * kk);     // B frag from LDS, used twice
            c0 = __builtin_amdgcn_wmma_f32_16x16x4_f32(
                     false, a0[kk], false, bf, (short)0, c0, false, false);
            c1 = __builtin_amdgcn_wmma_f32_16x16x4_f32(
                     false, a1[kk], false, bf, (short)0, c1, false, false);
        }
#pragma unroll
        for (int r = 0; r < 8; ++r) {
            m0[r] = fmaxf(m0[r], c0[r]);
            m1[r] = fmaxf(m1[r], c1[r]);
        }
    }

    // init_scores at offset 0, prev_scores at offset B*2*HW; channel = set&1
    float* dst = out + (size_t)(set >> 1) * (BB * 2 * HW)
                     + ((size_t)b * 2 + (set & 1)) * HW + p0;

    // Row M=r: lanes 0-15 hold pixels p0+r, lanes 16-31 hold p0+8+r (m0),
    // +16 for m1. Max-reduce each 16-lane group, lanes 0/16 store.
#pragma unroll
    for (int r = 0; r < 8; ++r) {
        float v0 = m0[r], v1 = m1[r];
#pragma unroll
        for (int off = 8; off > 0; off >>= 1) {
            v0 = fmaxf(v0, __shfl_xor(v0, off, 32));
            v1 = fmaxf(v1, __shfl_xor(v1, off, 32));
        }
        if (lane == 0)        { dst[r]      = v0; dst[16 + r] = v1; }
        else if (lane == 16)  { dst[8 + r]  = v0; dst[24 + r] = v1; }
    }
}

// ---------------------------------------------------------------------------
extern "C" void kernel_launch(void* const* d_in, const int* in_sizes, int n_in,
                              void* d_out, int out_size, void* d_ws, size_t ws_size,
                              hipStream_t stream) {
    const float* feats      = (const float*)d_in[0];  // [B, CIN, H, W]
    const float* conv_w     = (const float*)d_in[1];  // [COUT, CIN]
    const float* conv_b     = (const float*)d_in[2];  // [COUT]
    const float* pm_init_bg = (const float*)d_in[3];  // [B, N, COUT]
    const float* pm_init_fg = (const float*)d_in[4];
    const float* pm_prev_bg = (const float*)d_in[5];
    const float* pm_prev_fg = (const float*)d_in[6];
    const float* pm_int_bg  = (const float*)d_in[7];  // [B, 1, COUT]
    const float* pm_int_fg  = (const float*)d_in[8];

    float* out = (float*)d_out;
    // Output layout (flat, return order):
    //   init_scores [B,2,HW] | prev_scores [B,2,HW] | int_scores [B,2,HW] | conv_feats_norm [B,COUT,HW]
    float* out_init = out;
    float* out_int  = out + 2 * (size_t)BB * 2 * HW;
    float* out_norm = out + 3 * (size_t)BB * 2 * HW;

    float* convT = (float*)d_ws;  // [B, HW, COUT] = 8 MB scratch

    // 1) conv GEMM -> transposed unnormalized features (feats single-pass)
    plm_conv_gemm<<<dim3(HW / 16, BB), dim3(32), 0, stream>>>(
        feats, conv_w, conv_b, convT);

    // 2) L2 normalize (in place) + scatter norm output + pm_int scores
    plm_normalize<<<dim3((BB * HW) / 8), dim3(256), 0, stream>>>(
        convT, pm_int_bg, pm_int_fg, out_norm, out_int);

    // 3) four correlation GEMMs with LDS-staged templates + fused max
    plm_corr_max<<<dim3(HW / 256, 4, BB), dim3(256), 0, stream>>>(
        convT, pm_init_bg, pm_init_fg, pm_prev_bg, pm_prev_fg, out_init);
}